// PredictYOLO_v2_37460704755795
// MI455X (gfx1250) — compile-verified
//
#include <hip/hip_runtime.h>
#include <cstdint>

typedef float v2f __attribute__((ext_vector_type(2)));
typedef float v8f __attribute__((ext_vector_type(8)));

#define NUM_CLASSES 20
#define NUM_ANC     5
#define GRID_W      52
#define HW          2704              // 52*52
#define NBATCH      64
#define NPB         (HW * NUM_ANC)    // 13520 candidates per batch
#define N_ALL       (NBATCH * NPB)    // 865280 total candidates
#define BATCH_STRIDE (125 * HW)       // 338000 floats per batch image
#define K_TOP       2048
#define CONF_THRF   0.35f
#define NMS_THRF    0.5f
#define CAP         8192              // max candidates kept for final sort
#define NBINS       1024
#define BIN_SHIFT   14
#define BIN_BASE    64192u            // 0x3EB00000 >> 14  (float bits of 0.34375)

// ---- workspace byte offsets (total ~21.4 MB) ----
#define OFF_SCORES  0u                       // N_ALL floats
#define OFF_BOXES   3461120u                 // N_ALL float4
#define OFF_LABELS  17305600u                // N_ALL int
#define OFF_HIST    20766720u                // NBINS u32
#define OFF_CNT     20770816u                // 16 u32 : [0]=candCount [1]=minKeyBits
#define OFF_CAND    20770880u                // CAP u64
#define OFF_NMSB    20836416u                // K_TOP float4 (offset boxes for NMS)
#define OFF_AREA    20869184u                // K_TOP floats
#define OFF_SUP     20877376u                // K_TOP*64 u32 suppression bit-matrix

// ---- d_out float offsets (ids, boxes, labels, scores, keep) ----
#define OUT_IDS     0
#define OUT_BOXES   2048
#define OUT_LABELS  10240
#define OUT_SCORES  12288
#define OUT_KEEP    14336

__device__ __forceinline__ float sigm(float x) { return 1.0f / (1.0f + __expf(-x)); }
__device__ __forceinline__ float clamp01(float x) { return fminf(fmaxf(x, 0.0f), 1.0f); }

// ---------------- 1) decode all candidates ----------------
__global__ __launch_bounds__(256) void k_decode(const float* __restrict__ pyolos,
                                                const float* __restrict__ ancs,
                                                float* __restrict__ scores,
                                                float* __restrict__ boxes,
                                                int* __restrict__ labels) {
    int tid = blockIdx.x * blockDim.x + threadIdx.x;
    if (tid >= N_ALL) return;
    int b   = tid / NPB;
    int r   = tid - b * NPB;
    int a   = r / HW;
    int pos = r - a * HW;
    const float* base = pyolos + (size_t)b * BATCH_STRIDE + (size_t)a * HW + pos;

    float conf = base[0];
    float maxl = base[(size_t)(1 * NUM_ANC) * HW];
    int arg = 0;
#pragma unroll
    for (int f = 2; f <= NUM_CLASSES; ++f) {
        float v = base[(size_t)(f * NUM_ANC) * HW];
        if (v > maxl) { maxl = v; arg = f - 1; }
    }
    float tx = base[(size_t)(21 * NUM_ANC) * HW];
    float ty = base[(size_t)(22 * NUM_ANC) * HW];
    float tw = base[(size_t)(23 * NUM_ANC) * HW];
    float th = base[(size_t)(24 * NUM_ANC) * HW];

    float score = sigm(conf) * sigm(maxl);
    float gx = (float)(pos % GRID_W);
    float gy = (float)(pos / GRID_W);
    float cx = (sigm(tx) + gx) * (1.0f / 52.0f);
    float cy = (sigm(ty) + gy) * (1.0f / 52.0f);
    float pw = __expf(tw) * ancs[2 * a + 0];
    float ph = __expf(th) * ancs[2 * a + 1];
    float l = clamp01(cx - 0.5f * pw);
    float t = clamp01(cy - 0.5f * ph);
    float rr = clamp01(cx + 0.5f * pw);
    float bb = clamp01(cy + 0.5f * ph);

    int g = b * NPB + pos * NUM_ANC + a;   // raveled index matching reference order
    scores[g] = score;
    ((float4*)boxes)[g] = make_float4(l, t, rr, bb);
    labels[g] = arg + 1;
}

// ---------------- 2) histogram of score float-bits ----------------
__global__ __launch_bounds__(256) void k_hist(const float* __restrict__ scores,
                                              unsigned* __restrict__ hist) {
    __shared__ unsigned lh[NBINS];
    for (int i = threadIdx.x; i < NBINS; i += blockDim.x) lh[i] = 0u;
    __syncthreads();
    int tid = blockIdx.x * blockDim.x + threadIdx.x;
    if (tid < N_ALL) {
        float s = scores[tid];
        if (s > CONF_THRF) {
            unsigned bin = (__float_as_uint(s) >> BIN_SHIFT) - BIN_BASE;
            if (bin > NBINS - 1u) bin = NBINS - 1u;
            atomicAdd(&lh[bin], 1u);
        }
    }
    __syncthreads();
    for (int i = threadIdx.x; i < NBINS; i += blockDim.x)
        if (lh[i]) atomicAdd(&hist[i], lh[i]);
}

// ---------------- 3) find threshold bin ----------------
__global__ void k_thresh(const unsigned* __restrict__ hist, unsigned* __restrict__ cnt) {
    unsigned acc = 0;
    int bin = NBINS - 1;
    for (; bin >= 0; --bin) { acc += hist[bin]; if (acc >= K_TOP) break; }
    if (bin < 0) bin = 0;
    cnt[1] = ((unsigned)bin + BIN_BASE) << BIN_SHIFT;   // lower-bound bit pattern
}

// ---------------- 4) compact survivors ----------------
__global__ __launch_bounds__(256) void k_compact(const float* __restrict__ scores,
                                                 unsigned* __restrict__ cnt,
                                                 unsigned long long* __restrict__ cand) {
    int tid = blockIdx.x * blockDim.x + threadIdx.x;
    if (tid >= N_ALL) return;
    float s = scores[tid];
    if (s > CONF_THRF) {
        unsigned bits = __float_as_uint(s);
        if (bits >= cnt[1]) {
            unsigned pos = atomicAdd(&cnt[0], 1u);
            if (pos < CAP)  // key: score bits desc, then smaller index first
                cand[pos] = ((unsigned long long)bits << 32) | (0xFFFFFFFFu - (unsigned)tid);
        }
    }
}

// ---------------- 5) bitonic sort + emit top-K ----------------
__global__ __launch_bounds__(1024) void k_sort_emit(const unsigned long long* __restrict__ cand,
                                                    const unsigned* __restrict__ cnt,
                                                    const float* __restrict__ boxes,
                                                    const int* __restrict__ labels,
                                                    float* __restrict__ out,
                                                    float* __restrict__ nmsb,
                                                    float* __restrict__ area) {
    extern __shared__ unsigned long long skey[];   // CAP * 8 = 64 KB dynamic LDS
    unsigned n = cnt[0]; if (n > CAP) n = CAP;
    for (int i = threadIdx.x; i < CAP; i += 1024) skey[i] = (i < (int)n) ? cand[i] : 0ULL;
    __syncthreads();
    for (int k = 2; k <= CAP; k <<= 1) {
        for (int j = k >> 1; j > 0; j >>= 1) {
#pragma unroll
            for (int t = 0; t < CAP / 1024; ++t) {
                int i = threadIdx.x + t * 1024;
                int ixj = i ^ j;
                if (ixj > i) {
                    unsigned long long a = skey[i], b = skey[ixj];
                    bool sw = ((i & k) == 0) ? (a < b) : (a > b);  // descending
                    if (sw) { skey[i] = b; skey[ixj] = a; }
                }
            }
            __syncthreads();
        }
    }
    for (int k = threadIdx.x; k < K_TOP; k += 1024) {
        unsigned long long key = skey[k];
        float score = __uint_as_float((unsigned)(key >> 32));
        unsigned g = 0xFFFFFFFFu - (unsigned)key;
        if (key == 0ULL) { g = 0u; score = 0.0f; }
        int id = (int)(g / NPB);
        int lab = labels[g];
        float4 bx = ((const float4*)boxes)[g];
        out[OUT_IDS + k]    = (float)id;
        out[OUT_BOXES + 4 * k + 0] = bx.x;
        out[OUT_BOXES + 4 * k + 1] = bx.y;
        out[OUT_BOXES + 4 * k + 2] = bx.z;
        out[OUT_BOXES + 4 * k + 3] = bx.w;
        out[OUT_LABELS + k] = (float)lab;
        out[OUT_SCORES + k] = score;
        float off = (float)(id * (NUM_CLASSES + 1) + lab) * 2.0f;
        ((float4*)nmsb)[k] = make_float4(bx.x + off, bx.y + off, bx.z + off, bx.w + off);
        area[k] = (bx.z - bx.x) * (bx.w - bx.y);
    }
}

// ---------------- 6) pairwise IoU>thr bit-matrix; area-sum via WMMA ----------------
// Per wave: one 16x16 tile.  S = A(16x4) x B(4x16) with A=[area_i,1,0,0], B=[1;area_j;0;0]
// so D[i][j] = area_i + area_j via v_wmma_f32_16x16x4_f32.  Intersection in VALU.
__global__ __launch_bounds__(256) void k_iou(const float* __restrict__ nmsb,
                                             const float* __restrict__ area,
                                             unsigned* __restrict__ sup) {
    int wave = threadIdx.x >> 5;
    int lane = threadIdx.x & 31;
    int tile = blockIdx.x * 8 + wave;          // 2048 blocks * 8 waves = 128*128 tiles
    int ib = (tile >> 7) << 4;
    int jb = (tile & 127) << 4;
    int l15 = lane & 15;
    bool lo = lane < 16;

    float ai = area[ib + l15];
    float aj = area[jb + l15];
    v2f A, Bm;
    A.x  = lo ? ai : 0.0f;    // A: K=0 (lanes 0-15) / K=2 (lanes 16-31, zero)
    A.y  = lo ? 1.0f : 0.0f;  // A: K=1 / K=3
    Bm.x = lo ? 1.0f : 0.0f;  // B row K=0 = ones / K=2 zero
    Bm.y = lo ? aj : 0.0f;    // B row K=1 = area_j / K=3 zero
    v8f C = {};
    v8f D = __builtin_amdgcn_wmma_f32_16x16x4_f32(false, A, false, Bm, (short)0, C,
                                                  false, false);
#pragma unroll
    for (int r = 0; r < 8; ++r) {
        int i = ib + r + (lo ? 0 : 8);         // D layout: lanes 0-15 -> M=r, 16-31 -> M=r+8
        int j = jb + l15;
        float4 bi = ((const float4*)nmsb)[i];
        float4 bj = ((const float4*)nmsb)[j];
        float w  = fmaxf(fminf(bi.z, bj.z) - fmaxf(bi.x, bj.x), 0.0f);
        float h  = fmaxf(fminf(bi.w, bj.w) - fmaxf(bi.y, bj.y), 0.0f);
        float inter = w * h;
        float ssum = D[r];                     // area_i + area_j from WMMA
        bool over = inter > NMS_THRF * (ssum - inter + 1e-9f);
        unsigned m = (unsigned)__ballot(over); // wave32: [15:0]=row r, [31:16]=row r+8
        if (lane == 0) {
            int word = jb >> 5, shift = jb & 16;
            unsigned lowb = m & 0xFFFFu, highb = (m >> 16) & 0xFFFFu;
            if (lowb)  atomicOr(&sup[(ib + r) * 64 + word],     lowb  << shift);
            if (highb) atomicOr(&sup[(ib + r + 8) * 64 + word], highb << shift);
        }
    }
}

// ---------------- 7) single-wave sequential greedy NMS on bit rows ----------------
__global__ __launch_bounds__(32) void k_nms(const unsigned* __restrict__ sup,
                                            float* __restrict__ out) {
    __shared__ unsigned keepW[64];
    int lane = threadIdx.x;
    for (int w = lane; w < 64; w += 32) {
        unsigned bits = 0;
        for (int t = 0; t < 32; ++t)
            if (out[OUT_SCORES + w * 32 + t] > CONF_THRF) bits |= (1u << t);
        keepW[w] = bits;
    }
    __syncthreads();
    for (int i = 0; i < K_TOP; ++i) {
        unsigned kb = keepW[i >> 5];
        if ((kb >> (i & 31)) & 1u) {
#pragma unroll
            for (int t = 0; t < 2; ++t) {
                int w = lane + t * 32;
                int wbase = w << 5;
                unsigned wm;
                if (wbase + 31 <= i)      wm = 0u;                       // all j <= i
                else if (wbase > i)       wm = 0xFFFFFFFFu;              // all j > i
                else                      wm = 0xFFFFFFFFu << ((i & 31) + 1);
                unsigned sv = sup[i * 64 + w] & wm;
                if (sv) keepW[w] &= ~sv;
            }
        }
    }
    for (int w = lane; w < 64; w += 32) {
        unsigned bits = keepW[w];
        for (int t = 0; t < 32; ++t)
            out[OUT_KEEP + w * 32 + t] = (bits >> t) & 1u ? 1.0f : 0.0f;
    }
}

extern "C" void kernel_launch(void* const* d_in, const int* in_sizes, int n_in,
                              void* d_out, int out_size, void* d_ws, size_t ws_size,
                              hipStream_t stream) {
    const float* pyolos = (const float*)d_in[0];
    const float* ancs   = (const float*)d_in[1];
    float* out = (float*)d_out;
    char*  ws  = (char*)d_ws;

    float*    scores = (float*)(ws + OFF_SCORES);
    float*    boxes  = (float*)(ws + OFF_BOXES);
    int*      labels = (int*)(ws + OFF_LABELS);
    unsigned* hist   = (unsigned*)(ws + OFF_HIST);
    unsigned* cnt    = (unsigned*)(ws + OFF_CNT);
    unsigned long long* cand = (unsigned long long*)(ws + OFF_CAND);
    float*    nmsb   = (float*)(ws + OFF_NMSB);
    float*    area   = (float*)(ws + OFF_AREA);
    unsigned* sup    = (unsigned*)(ws + OFF_SUP);

    hipMemsetAsync(ws + OFF_HIST, 0, NBINS * 4 + 64, stream);          // hist + counters
    hipMemsetAsync(ws + OFF_SUP,  0, K_TOP * 64 * 4, stream);          // suppression bits

    k_decode <<<N_ALL / 256, 256, 0, stream>>>(pyolos, ancs, scores, boxes, labels);
    k_hist   <<<N_ALL / 256, 256, 0, stream>>>(scores, hist);
    k_thresh <<<1, 1, 0, stream>>>(hist, cnt);
    k_compact<<<N_ALL / 256, 256, 0, stream>>>(scores, cnt, cand);
    k_sort_emit<<<1, 1024, CAP * sizeof(unsigned long long), stream>>>(cand, cnt, boxes,
                                                                       labels, out, nmsb, area);
    k_iou    <<<(128 * 128) / 8, 256, 0, stream>>>(nmsb, area, sup);
    k_nms    <<<1, 32, 0, stream>>>(sup, out);
}